// ResidualBlock_41918880809282
// MI455X (gfx1250) — compile-verified
//
#include <hip/hip_runtime.h>
#include <hip/hip_bf16.h>

#define CH 128
#define EPSV 1e-5f

typedef __attribute__((ext_vector_type(2))) float v2f;
typedef __attribute__((ext_vector_type(8))) float v8f;

// ---------------- zero scratch (float4 fast path) ----------------
__global__ void zero_kernel(float* __restrict__ p, long long n) {
    long long n4 = n >> 2;
    long long i = (long long)blockIdx.x * blockDim.x + threadIdx.x;
    long long stride = (long long)gridDim.x * blockDim.x;
    float4 z = make_float4(0.f, 0.f, 0.f, 0.f);
    for (long long j = i; j < n4; j += stride) ((float4*)p)[j] = z;
    // scalar tail
    for (long long j = (n4 << 2) + i; j < n; j += stride) p[j] = 0.0f;
}

// ---------------- degree: deg[dst] += 1 ----------------
__global__ __launch_bounds__(256) void degree_kernel(const int* __restrict__ ei, float* __restrict__ degf, int E) {
    int i = blockIdx.x * blockDim.x + threadIdx.x;
    if (i >= E) return;
    int dst = ei[E + i];
    atomicAdd(&degf[dst], 1.0f);
}

// ---------------- dinv = rsqrt(deg+1), invdeg = 1/(deg+1) ----------------
__global__ __launch_bounds__(256) void dinv_kernel(const float* __restrict__ degf, float* __restrict__ dinv,
                                                   float* __restrict__ invdeg, int N) {
    int i = blockIdx.x * blockDim.x + threadIdx.x;
    if (i >= N) return;
    float d = degf[i] + 1.0f;
    dinv[i] = rsqrtf(d);
    invdeg[i] = 1.0f / d;
}

// ---------------- H = X @ W via V_WMMA_F32_16X16X4_F32 ----------------
// Block: 256 thr = 8 waves; block computes 128 rows x 128 cols.
// W (128x128 f32, 64KB) staged in LDS.
__global__ __launch_bounds__(256) void gemm128_wmma(const float* __restrict__ X, const float* __restrict__ W,
                                                    float* __restrict__ H, int N) {
    __shared__ float sW[CH * CH];
    {
        const float4* Wv = (const float4*)W;
        float4* sWv = (float4*)sW;
        for (int i = threadIdx.x; i < CH * CH / 4; i += 256) sWv[i] = Wv[i];
    }
    __syncthreads();

    const int wave = threadIdx.x >> 5;
    const int lane = threadIdx.x & 31;
    const int half = lane >> 4;   // selects K pair (A) / K rows (B)
    const int l16  = lane & 15;

    const int rowBase = blockIdx.x * 128 + wave * 16;   // wave's 16 rows
    int m = rowBase + l16;
    int mc = m < N ? m : (N - 1);         // clamp: extra rows never stored
    const float* xrow = X + (size_t)mc * CH;

    v8f acc[8];
#pragma unroll
    for (int t = 0; t < 8; ++t) acc[t] = (v8f){0.f,0.f,0.f,0.f,0.f,0.f,0.f,0.f};

    for (int kk = 0; kk < CH / 4; ++kk) {
        const int ka = kk * 4 + half * 2;
        // A frag (16x4 f32): lane%16 = M row, VGPR0/1 = K = ka, ka+1
        v2f a;
        a.x = xrow[ka];
        a.y = xrow[ka + 1];
#pragma unroll
        for (int nt = 0; nt < 8; ++nt) {
            const int n = nt * 16 + l16;
            // B frag (4x16 f32): row-striped across lanes; VGPR0/1 = K = ka, ka+1
            v2f b;
            b.x = sW[ka * CH + n];
            b.y = sW[(ka + 1) * CH + n];
            acc[nt] = __builtin_amdgcn_wmma_f32_16x16x4_f32(
                false, a, false, b, (short)0, acc[nt], false, false);
        }
    }

    // C/D layout: VGPR r -> M=r (lanes 0-15) / M=r+8 (lanes 16-31); N = lane%16
    if (rowBase + 15 < N) {
        // interior fast path: wave-uniform, no per-store exec masking
        float* hbase = H + (size_t)(rowBase + half * 8) * CH + l16;
#pragma unroll
        for (int nt = 0; nt < 8; ++nt) {
            float* hcol = hbase + nt * 16;
#pragma unroll
            for (int r = 0; r < 8; ++r) {
                hcol[(size_t)r * CH] = acc[nt][r];
            }
        }
    } else {
        // tail block: guard each row
#pragma unroll
        for (int nt = 0; nt < 8; ++nt) {
            const int col = nt * 16 + l16;
#pragma unroll
            for (int r = 0; r < 8; ++r) {
                const int row = rowBase + r + half * 8;
                if (row < N) H[(size_t)row * CH + col] = acc[nt][r];
            }
        }
    }
}

// ---------------- edge scatter: agg[dst] += h[src] * dinv[src]*dinv[dst] ----------------
// One wave32 per edge; lane handles 4 channels (float4 gather, 4 f32 atomics).
// h and agg (51 MB each) are L2-resident on MI455X (192 MB L2).
__global__ __launch_bounds__(256) void scatter_kernel(const int* __restrict__ ei, const float* __restrict__ h,
                                                      const float* __restrict__ dinv, float* __restrict__ agg, int E) {
    int gid = blockIdx.x * blockDim.x + threadIdx.x;
    int e = gid >> 5;
    int lane = gid & 31;
    if (e >= E) return;
    int src = ei[e];
    int dst = ei[E + e];
    float norm = dinv[src] * dinv[dst];
    float4 v = ((const float4*)(h + (size_t)src * CH))[lane];
    float* ad = agg + (size_t)dst * CH + lane * 4;
    atomicAdd(ad + 0, v.x * norm);
    atomicAdd(ad + 1, v.y * norm);
    atomicAdd(ad + 2, v.z * norm);
    atomicAdd(ad + 3, v.w * norm);
}

// ---------------- finalize y = agg + h*invdeg + b (in place), accumulate per-channel sums ----------------
__global__ __launch_bounds__(256) void finalize_stats(float* __restrict__ agg, const float* __restrict__ h,
                                                      const float* __restrict__ invdeg, const float* __restrict__ b,
                                                      float* __restrict__ gsum, float* __restrict__ gsq, int N) {
    __shared__ float s_sum[CH];
    __shared__ float s_sq[CH];
    if (threadIdx.x < CH) { s_sum[threadIdx.x] = 0.f; s_sq[threadIdx.x] = 0.f; }
    __syncthreads();

    int gid = blockIdx.x * blockDim.x + threadIdx.x;  // one float4 per thread
    int total = N * (CH / 4);
    if (gid < total) {
        int node = gid >> 5;
        int q = gid & 31;
        int c0 = q * 4;
        float id = invdeg[node];
        float4 a = ((float4*)agg)[gid];
        float4 hv = ((const float4*)h)[gid];
        float4 bv = ((const float4*)b)[q];
        float4 y;
        y.x = a.x + hv.x * id + bv.x;
        y.y = a.y + hv.y * id + bv.y;
        y.z = a.z + hv.z * id + bv.z;
        y.w = a.w + hv.w * id + bv.w;
        ((float4*)agg)[gid] = y;
        atomicAdd(&s_sum[c0 + 0], y.x); atomicAdd(&s_sq[c0 + 0], y.x * y.x);
        atomicAdd(&s_sum[c0 + 1], y.y); atomicAdd(&s_sq[c0 + 1], y.y * y.y);
        atomicAdd(&s_sum[c0 + 2], y.z); atomicAdd(&s_sq[c0 + 2], y.z * y.z);
        atomicAdd(&s_sum[c0 + 3], y.w); atomicAdd(&s_sq[c0 + 3], y.w * y.w);
    }
    __syncthreads();
    if (threadIdx.x < CH) {
        atomicAdd(&gsum[threadIdx.x], s_sum[threadIdx.x]);
        atomicAdd(&gsq[threadIdx.x], s_sq[threadIdx.x]);
    }
}

// ---------------- BN scale/shift from stats ----------------
__global__ void bn_final(const float* __restrict__ gsum, const float* __restrict__ gsq,
                         const float* __restrict__ gamma, const float* __restrict__ beta,
                         float* __restrict__ scale, float* __restrict__ shift, int N) {
    int c = threadIdx.x;
    if (c >= CH) return;
    float invN = 1.0f / (float)N;
    float mean = gsum[c] * invN;
    float var = gsq[c] * invN - mean * mean;
    float rstd = rsqrtf(var + EPSV);
    float s = rstd * gamma[c];
    scale[c] = s;
    shift[c] = beta[c] - mean * s;
}

// ---------------- out = relu(relu(y*scale+shift) + x) ----------------
__global__ __launch_bounds__(256) void epilogue_kernel(const float* __restrict__ y, const float* __restrict__ x,
                                                       const float* __restrict__ scale, const float* __restrict__ shift,
                                                       float* __restrict__ out, int N) {
    int gid = blockIdx.x * blockDim.x + threadIdx.x;
    int total = N * (CH / 4);
    if (gid >= total) return;
    int q = gid & 31;
    float4 yv = ((const float4*)y)[gid];
    float4 xv = ((const float4*)x)[gid];
    float4 sc = ((const float4*)scale)[q];
    float4 sh = ((const float4*)shift)[q];
    float4 o;
    o.x = fmaxf(fmaxf(yv.x * sc.x + sh.x, 0.f) + xv.x, 0.f);
    o.y = fmaxf(fmaxf(yv.y * sc.y + sh.y, 0.f) + xv.y, 0.f);
    o.z = fmaxf(fmaxf(yv.z * sc.z + sh.z, 0.f) + xv.z, 0.f);
    o.w = fmaxf(fmaxf(yv.w * sc.w + sh.w, 0.f) + xv.w, 0.f);
    ((float4*)out)[gid] = o;
}

extern "C" void kernel_launch(void* const* d_in, const int* in_sizes, int n_in,
                              void* d_out, int out_size, void* d_ws, size_t ws_size,
                              hipStream_t stream) {
    const float* x     = (const float*)d_in[0];
    const int*   ei    = (const int*)d_in[1];
    const float* W     = (const float*)d_in[2];
    const float* b     = (const float*)d_in[3];
    const float* gamma = (const float*)d_in[4];
    const float* beta  = (const float*)d_in[5];
    float* out = (float*)d_out;

    const int N = in_sizes[0] / CH;
    const int E = in_sizes[1] / 2;

    // workspace layout (floats)
    float* ws     = (float*)d_ws;
    float* h      = ws;                                 // N*CH
    float* agg    = h + (size_t)N * CH;                 // N*CH (zeroed; becomes y)
    float* degf   = agg + (size_t)N * CH;               // N    (zeroed)
    float* gsum   = degf + N;                           // CH   (zeroed)
    float* gsq    = gsum + CH;                          // CH   (zeroed)
    float* dinv   = gsq + CH;                           // N
    float* invdeg = dinv + N;                           // N
    float* scale  = invdeg + N;                         // CH
    float* shift  = scale + CH;                         // CH

    // 1) zero accumulated regions: agg .. gsq (contiguous)
    long long nz = (long long)N * CH + N + 2 * CH;
    zero_kernel<<<4096, 256, 0, stream>>>(agg, nz);

    // 2) degree
    degree_kernel<<<(E + 255) / 256, 256, 0, stream>>>(ei, degf, E);

    // 3) dinv / invdeg
    dinv_kernel<<<(N + 255) / 256, 256, 0, stream>>>(degf, dinv, invdeg, N);

    // 4) h = x @ W  (WMMA f32 16x16x4)
    gemm128_wmma<<<(N + 127) / 128, 256, 0, stream>>>(x, W, h, N);

    // 5) edge scatter-add (wave per edge)
    {
        long long threads = (long long)E * 32;
        int blocks = (int)((threads + 255) / 256);
        scatter_kernel<<<blocks, 256, 0, stream>>>(ei, h, dinv, agg, E);
    }

    // 6) finalize y + batch statistics
    {
        int total = N * (CH / 4);
        finalize_stats<<<(total + 255) / 256, 256, 0, stream>>>(agg, h, invdeg, b, gsum, gsq, N);
    }

    // 7) BN scale/shift
    bn_final<<<1, CH, 0, stream>>>(gsum, gsq, gamma, beta, scale, shift, N);

    // 8) epilogue: BN apply + ReLU + residual + ReLU
    {
        int total = N * (CH / 4);
        epilogue_kernel<<<(total + 255) / 256, 256, 0, stream>>>(agg, x, scale, shift, out, N);
    }
}